// TransOp_expm_14164802142649
// MI455X (gfx1250) — compile-verified
//
#include <hip/hip_runtime.h>
#include <hip/hip_bf16.h>
#include <stdint.h>

// Batched 3x3 matrix exponential (scaling-and-squaring, order-12 Horner Taylor)
// applied to a 3x1 vector:  out[b] = expm(sum_m c[b,m] * psi[m]) @ x[b]
//
// B = 2,097,152 batches, N = 3, M = 6.
// Strategy: one batch per lane (wave32), everything in registers, fully
// unrolled. psi (54 uniform floats) staged to LDS with the gfx1250 async
// global->LDS copy (ASYNCcnt) once per workgroup.

#define SQUARINGS 8
#define ORDER 12

__device__ __forceinline__ void mm3(float* __restrict__ D,
                                    const float* __restrict__ A,
                                    const float* __restrict__ Bm) {
#pragma unroll
  for (int i = 0; i < 3; ++i) {
#pragma unroll
    for (int j = 0; j < 3; ++j) {
      float s = A[i * 3 + 0] * Bm[0 * 3 + j];
      s = fmaf(A[i * 3 + 1], Bm[1 * 3 + j], s);
      s = fmaf(A[i * 3 + 2], Bm[2 * 3 + j], s);
      D[i * 3 + j] = s;
    }
  }
}

__global__ __launch_bounds__(256) void expm3_apply_kernel(
    const float* __restrict__ x,    // [B,3]
    const float* __restrict__ c,    // [B,6]
    const float* __restrict__ psi,  // [6,3,3] = 54 floats
    float* __restrict__ out,        // [B,3]
    int B) {
  __shared__ float s_psi[54];

  const int t = threadIdx.x;

  // ---- Stage psi into LDS via gfx1250 async global->LDS copy (ASYNCcnt). ----
  // Lanes 0..53 each move one dword. The generic pointer to a __shared__
  // object carries the LDS byte offset in its low 32 bits (LDS aperture
  // truncation, ISA 10.2), which is exactly what VDST of
  // global_load_async_to_lds_b32 wants.
  if (t < 54) {
    unsigned lds_off = (unsigned)(uintptr_t)(&s_psi[t]);
    unsigned long long gaddr = (unsigned long long)(uintptr_t)(psi + t);
    asm volatile("global_load_async_to_lds_b32 %0, %1, off"
                 :
                 : "v"(lds_off), "v"(gaddr)
                 : "memory");
  }
#if __has_builtin(__builtin_amdgcn_s_wait_asynccnt)
  __builtin_amdgcn_s_wait_asynccnt(0);
#else
  asm volatile("s_wait_asynccnt 0" ::: "memory");
#endif
  __syncthreads();

  const long long b = (long long)blockIdx.x * blockDim.x + t;
  if (b >= B) return;

  // ---- Load per-batch coefficients. ----
  float cm[6];
#pragma unroll
  for (int m = 0; m < 6; ++m) cm[m] = c[b * 6 + m];

  // ---- T = sum_m c[b,m] * psi[m]  (3x3, registers). ----
  float T[9];
#pragma unroll
  for (int j = 0; j < 9; ++j) {
    float s = cm[0] * s_psi[0 * 9 + j];
#pragma unroll
    for (int m = 1; m < 6; ++m) s = fmaf(cm[m], s_psi[m * 9 + j], s);
    T[j] = s;
  }

  // ---- A = T / 2^SQUARINGS. ----
  const float scale = 1.0f / (float)(1 << SQUARINGS);  // 2^-8, exact
  float A[9];
#pragma unroll
  for (int j = 0; j < 9; ++j) A[j] = T[j] * scale;

  // ---- Horner Taylor: E = I + A/ORDER; E = I + (A/k) @ E, k=ORDER-1..1. ----
  float E[9];
  {
    const float inv = 1.0f / (float)ORDER;
#pragma unroll
    for (int i = 0; i < 3; ++i)
#pragma unroll
      for (int j = 0; j < 3; ++j)
        E[i * 3 + j] = fmaf(A[i * 3 + j], inv, (i == j) ? 1.0f : 0.0f);
  }
#pragma unroll
  for (int k = ORDER - 1; k >= 1; --k) {
    float P[9];
    mm3(P, A, E);                       // (A @ E)
    const float invk = 1.0f / (float)k; // constant-folded per unrolled k
#pragma unroll
    for (int i = 0; i < 3; ++i)
#pragma unroll
      for (int j = 0; j < 3; ++j)
        E[i * 3 + j] = fmaf(P[i * 3 + j], invk, (i == j) ? 1.0f : 0.0f);
  }

  // ---- Repeated squaring: E <- E @ E, SQUARINGS times. ----
#pragma unroll
  for (int s = 0; s < SQUARINGS; ++s) {
    float P[9];
    mm3(P, E, E);
#pragma unroll
    for (int j = 0; j < 9; ++j) E[j] = P[j];
  }

  // ---- y = E @ x[b]. ----
  float xv[3];
#pragma unroll
  for (int j = 0; j < 3; ++j) xv[j] = x[b * 3 + j];

#pragma unroll
  for (int i = 0; i < 3; ++i) {
    float s = E[i * 3 + 0] * xv[0];
    s = fmaf(E[i * 3 + 1], xv[1], s);
    s = fmaf(E[i * 3 + 2], xv[2], s);
    out[b * 3 + i] = s;
  }
}

extern "C" void kernel_launch(void* const* d_in, const int* in_sizes, int n_in,
                              void* d_out, int out_size, void* d_ws, size_t ws_size,
                              hipStream_t stream) {
  const float* x = (const float*)d_in[0];    // [B,3,1]
  const float* c = (const float*)d_in[1];    // [B,6]
  const float* psi = (const float*)d_in[2];  // [6,3,3]
  float* out = (float*)d_out;                // [B,3,1]

  const int B = in_sizes[0] / 3;
  const int block = 256;
  const int grid = (B + block - 1) / block;

  expm3_apply_kernel<<<grid, block, 0, stream>>>(x, c, psi, out, B);
}